// SpatialTransformer_82222853914715
// MI455X (gfx1250) — compile-verified
//
#include <hip/hip_runtime.h>

// ---------------------------------------------------------------------------
// 3D spatial transformer (trilinear warp), B=2, C=1, vol = 160x192x224, f32.
// Memory-bound gather: ~275 MB effective traffic -> ~12 us floor at 23.3 TB/s.
// Strategy: NT-stream flow/out (keep 55 MB src resident in 192 MB L2),
// 4-wide x-vectorized b128 loads/stores, branchless masked trilinear blend.
// ---------------------------------------------------------------------------

#define DD 160
#define HH 192
#define WW 224
constexpr int VOL = DD * HH * WW;   // 6,881,280
constexpr int HW  = HH * WW;        // 43,008
constexpr int NB  = 2;

typedef __attribute__((ext_vector_type(4))) float v4f;

__device__ __forceinline__ float sample_one(const float* __restrict__ s,
                                            float cz, float cy, float cx) {
    float zf = floorf(cz), yf = floorf(cy), xf = floorf(cx);
    int z0 = (int)zf, y0 = (int)yf, x0 = (int)xf;
    float tz = cz - zf, ty = cy - yf, tx = cx - xf;

    // Per-axis weights with validity folded in: product == wz*wy*wx*valid.
    float az0 = (z0     >= 0 && z0     < DD) ? (1.0f - tz) : 0.0f;
    float az1 = (z0 + 1 >= 0 && z0 + 1 < DD) ? tz          : 0.0f;
    float ay0 = (y0     >= 0 && y0     < HH) ? (1.0f - ty) : 0.0f;
    float ay1 = (y0 + 1 >= 0 && y0 + 1 < HH) ? ty          : 0.0f;
    float ax0 = (x0     >= 0 && x0     < WW) ? (1.0f - tx) : 0.0f;
    float ax1 = (x0 + 1 >= 0 && x0 + 1 < WW) ? tx          : 0.0f;

    // Clamped indices for the actual loads (match jnp.clip in reference).
    int zc0 = min(max(z0,     0), DD - 1), zc1 = min(max(z0 + 1, 0), DD - 1);
    int yc0 = min(max(y0,     0), HH - 1), yc1 = min(max(y0 + 1, 0), HH - 1);
    int xc0 = min(max(x0,     0), WW - 1), xc1 = min(max(x0 + 1, 0), WW - 1);

    const float* p00 = s + zc0 * HW + yc0 * WW;
    const float* p01 = s + zc0 * HW + yc1 * WW;
    const float* p10 = s + zc1 * HW + yc0 * WW;
    const float* p11 = s + zc1 * HW + yc1 * WW;

    // 8 gathers (RT cache policy: src volume is L2-resident).
    float v000 = p00[xc0], v001 = p00[xc1];
    float v010 = p01[xc0], v011 = p01[xc1];
    float v100 = p10[xc0], v101 = p10[xc1];
    float v110 = p11[xc0], v111 = p11[xc1];

    float w00 = az0 * ay0, w01 = az0 * ay1;
    float w10 = az1 * ay0, w11 = az1 * ay1;

    float r = v000 * (w00 * ax0);
    r = fmaf(v001, w00 * ax1, r);
    r = fmaf(v010, w01 * ax0, r);
    r = fmaf(v011, w01 * ax1, r);
    r = fmaf(v100, w10 * ax0, r);
    r = fmaf(v101, w10 * ax1, r);
    r = fmaf(v110, w11 * ax0, r);
    r = fmaf(v111, w11 * ax1, r);
    return r;
}

__global__ __launch_bounds__(256)
void warp3d_kernel(const float* __restrict__ src,
                   const float* __restrict__ flow,
                   float* __restrict__ out) {
    constexpr int NQ = NB * VOL / 4;          // 3,440,640 quads
    int q = blockIdx.x * blockDim.x + threadIdx.x;
    if (q >= NQ) return;

    int base = q * 4;                          // flat voxel index, multiple of 4
    int b   = base / VOL;
    int r   = base - b * VOL;
    int z   = r / HW;
    int rem = r - z * HW;
    int y   = rem / WW;
    int x   = rem - y * WW;                    // multiple of 4 (WW % 4 == 0)

    // Flow: channel-major (dz, dy, dx planes). Stream with NT hint so the
    // 165 MB flow stream does not evict the L2-resident src volume.
    const float* fb = flow + (size_t)b * 3 * VOL + r;
    v4f fz = __builtin_nontemporal_load((const v4f*)(fb));
    v4f fy = __builtin_nontemporal_load((const v4f*)(fb + VOL));
    v4f fx = __builtin_nontemporal_load((const v4f*)(fb + 2 * VOL));

    const float* sb = src + (size_t)b * VOL;

    v4f res;
#pragma unroll
    for (int j = 0; j < 4; ++j) {
        float cz = (float)z       + fz[j];
        float cy = (float)y       + fy[j];
        float cx = (float)(x + j) + fx[j];
        res[j] = sample_one(sb, cz, cy, cx);
    }

    // Output written once: NT store (b128).
    __builtin_nontemporal_store(res, (v4f*)(out + base));
}

extern "C" void kernel_launch(void* const* d_in, const int* in_sizes, int n_in,
                              void* d_out, int out_size, void* d_ws, size_t ws_size,
                              hipStream_t stream) {
    const float* src  = (const float*)d_in[0];   // (2,1,160,192,224) f32
    const float* flow = (const float*)d_in[1];   // (2,3,160,192,224) f32
    float*       out  = (float*)d_out;           // (2,1,160,192,224) f32

    constexpr int NQ = NB * VOL / 4;
    dim3 block(256);                              // 8 wave32 waves / block
    dim3 grid((NQ + block.x - 1) / block.x);
    warp3d_kernel<<<grid, block, 0, stream>>>(src, flow, out);
}